// GCN_60112362275286
// MI455X (gfx1250) — compile-verified
//
#include <hip/hip_runtime.h>
#include <math.h>

typedef float v2f __attribute__((ext_vector_type(2)));
typedef float v8f __attribute__((ext_vector_type(8)));

#define NNODES 50000
#define EPSBN 1e-5f

// ---------------- small utility kernels ----------------
__global__ void k_fill(float* __restrict__ p, float v, int n) {
    int i = blockIdx.x * blockDim.x + threadIdx.x;
    if (i < n) p[i] = v;
}

__global__ void k_deg_scatter(const long long* __restrict__ ei, float* __restrict__ deg, int E) {
    int e = blockIdx.x * blockDim.x + threadIdx.x;
    if (e < E) {
        int d = (int)ei[(long long)E + e];
        atomicAdd(&deg[d], 1.0f);
    }
}

__global__ void k_rsqrt_inplace(float* __restrict__ p, int n) {
    int i = blockIdx.x * blockDim.x + threadIdx.x;
    if (i < n) {
        float v = p[i];
        p[i] = (v > 0.f) ? rsqrtf(v) : 0.f;
    }
}

// out[i,f] = xw[i,f]*dinv[i]^2 + bias[f]   (self-loop term + bias, also zero-initializes out)
__global__ void k_agg_init(const float* __restrict__ xw, const float* __restrict__ dinv,
                           const float* __restrict__ bias, float* __restrict__ out,
                           int N, int F) {
    long long total = (long long)N * F;
    for (long long t = (long long)blockIdx.x * blockDim.x + threadIdx.x; t < total;
         t += (long long)gridDim.x * blockDim.x) {
        int i = (int)(t / F);
        int f = (int)(t - (long long)i * F);
        float di = dinv[i];
        out[t] = xw[t] * di * di + bias[f];
    }
}

// scatter: out[dst,f] += xw[src,f]*dinv[src]*dinv[dst]
__global__ void k_agg_edges(const long long* __restrict__ ei, const float* __restrict__ xw,
                            const float* __restrict__ dinv, float* __restrict__ out,
                            int E, int F) {
    long long total = (long long)E * F;
    for (long long t = (long long)blockIdx.x * blockDim.x + threadIdx.x; t < total;
         t += (long long)gridDim.x * blockDim.x) {
        int e = (int)(t / F);
        int f = (int)(t - (long long)e * F);
        int s = (int)ei[e];
        int d = (int)ei[(long long)E + e];
        atomicAdd(&out[(long long)d * F + f], xw[(long long)s * F + f] * dinv[s] * dinv[d]);
    }
}

// ---------------- batch norm ----------------
// stats[f] = sum, stats[128+f] = sumsq   (stats must be zeroed first)
__global__ void k_bn_stats(const float* __restrict__ X, float* __restrict__ stats, int N, int F) {
    int f = threadIdx.x;
    if (f >= F) return;
    float s = 0.f, s2 = 0.f;
    for (int r = blockIdx.x; r < N; r += gridDim.x) {
        float v = X[(long long)r * F + f];
        s += v; s2 += v * v;
    }
    atomicAdd(&stats[f], s);
    atomicAdd(&stats[128 + f], s2);
}

// ss[f]=scale, ss[128+f]=shift
__global__ void k_bn_final(const float* __restrict__ stats, const float* __restrict__ g,
                           const float* __restrict__ be, float* __restrict__ ss, int N, int F) {
    int f = threadIdx.x;
    if (f >= F) return;
    float inv_n = 1.0f / (float)N;
    float mean = stats[f] * inv_n;
    float var  = stats[128 + f] * inv_n - mean * mean;
    float rinv = rsqrtf(var + EPSBN);
    float scale = g[f] * rinv;
    ss[f]       = scale;
    ss[128 + f] = be[f] - mean * scale;
}

__global__ void k_bn_apply(float* __restrict__ X, const float* __restrict__ ss,
                           int N, int F, int relu) {
    long long total = (long long)N * F;
    for (long long t = (long long)blockIdx.x * blockDim.x + threadIdx.x; t < total;
         t += (long long)gridDim.x * blockDim.x) {
        int f = (int)(t % F);
        float y = X[t] * ss[f] + ss[128 + f];
        if (relu) y = fmaxf(y, 0.f);
        X[t] = y;
    }
}

// ---------------- WMMA fp32 GEMM (compile-time shape, branchless tails) ----------------
// C[M,NC] = A[M,K] * B (+ bias0 + bias1).  TRANSB==0: B is [K,NC]; TRANSB==1: B is [NC,K].
// One wave (32 threads) per 16x16 output tile; grid.x = M/16 (M=50000 is a multiple of 16).
//
// Tail policy:
//  - column tail (col >= NC): addresses are clamped to column NC-1; the duplicate values land
//    in accumulator columns that are never stored, so NO masking is needed.
//  - K tail (ka >= K, only for K=50/30 last iteration): B value is multiplied by a {1,0} mask,
//    so the load stays unconditional (no branch) and a*0 == 0 exactly.
//  - lane = threadIdx.x & 31 pins half in {0,1} via known-bits, so all interior-iteration
//    bound checks constant-fold after full unroll -> pure load+wmma stream, EXEC all-ones.
template <int K, int NC, int TRANSB>
__global__ void k_wmma_gemm(const float* __restrict__ A, const float* __restrict__ B,
                            const float* __restrict__ bias0, const float* __restrict__ bias1,
                            float* __restrict__ C) {
    const int lane = threadIdx.x & 31;  // known range [0,31]
    const int half = lane >> 4;         // 0: K pair {k,k+1}, 1: {k+2,k+3}
    const int l16  = lane & 15;
    const int mt = blockIdx.x * 16;
    const int nt = blockIdx.y * 16;
    const int row = mt + l16;                    // always < M
    const int col = nt + l16;
    const bool c_ok = (col < NC);
    const int  colc = c_ok ? col : (NC - 1);     // clamped, always legal
    const long long rowK  = (long long)row * K;
    const long long colcK = (long long)colc * K;
    const int h2 = half * 2;

    v8f acc = {};
#pragma unroll
    for (int k0 = 0; k0 < K; k0 += 4) {
        const int ka = k0 + h2;                            // range [k0, k0+2]
        const bool k0ok = (ka     < K);                    // folds true unless tail iter
        const bool k1ok = (ka + 1 < K);
        const int  kc0  = k0ok ? ka     : (K - 1);
        const int  kc1  = k1ok ? ka + 1 : (K - 1);
        const float m0  = k0ok ? 1.f : 0.f;                // folds to 1.0 in interior
        const float m1  = k1ok ? 1.f : 0.f;
        v2f a, b;
        a.x = A[rowK + kc0];                               // contiguous pair -> b64 load
        a.y = A[rowK + kc1];
        if (TRANSB) {
            b.x = B[colcK + kc0] * m0;
            b.y = B[colcK + kc1] * m1;
        } else {
            b.x = B[(long long)kc0 * NC + colc] * m0;
            b.y = B[(long long)kc1 * NC + colc] * m1;
        }
        acc = __builtin_amdgcn_wmma_f32_16x16x4_f32(
            false, a, false, b, (short)0, acc, false, false);
    }

    if (c_ok) {
        float bb = 0.f;
        if (bias0) bb += bias0[col];
        if (bias1) bb += bias1[col];
#pragma unroll
        for (int r = 0; r < 8; r++) {
            int rrow = mt + half * 8 + r;   // C/D layout: VGPR r -> M = r (+8 for upper half-wave)
            C[(long long)rrow * NC + col] = acc[r] + bb;
        }
    }
}

// ---------------- sequential RNN scan (single wave32, barriers -> S_NOP) ----------------
// h_t = tanh(u_t + Whh * h_{t-1}); u already holds x@Wih^T + bih + bhh.
// Each lane owns 2 hidden units (50 = 32 + 18); Whh rows live in registers; h is broadcast
// through LDS. Next step's u row is prefetched before the dependency wait to hide latency
// on the serial chain.
__global__ void k_rnn_scan(const float* __restrict__ u, const float* __restrict__ Whh,
                           float* __restrict__ out, int N) {
    __shared__ float hprev[50];
    const int l  = threadIdx.x & 31;   // 0..31
    const int j0 = l;                  // always < 50
    const int j1 = 32 + l;
    const bool has1 = (j1 < 50);

    float w0[50], w1[50];
#pragma unroll
    for (int k = 0; k < 50; k++) w0[k] = Whh[j0 * 50 + k];
    if (has1) {
#pragma unroll
        for (int k = 0; k < 50; k++) w1[k] = Whh[j1 * 50 + k];
    }
    hprev[j0] = 0.f;
    if (has1) hprev[j1] = 0.f;
    __syncthreads();

    float u0 = u[j0];
    float u1 = has1 ? u[j1] : 0.f;
    for (int t = 0; t < N; t++) {
        float a0 = u0, a1 = u1;
#pragma unroll
        for (int k = 0; k < 50; k++) {
            float h = hprev[k];
            a0 += w0[k] * h;
            a1 += w1[k] * h;
        }
        float h0 = tanhf(a0);
        float h1 = tanhf(a1);
        out[(long long)t * 50 + j0] = h0;
        if (has1) out[(long long)t * 50 + j1] = h1;
        // prefetch next u row while the LDS update settles
        if (t + 1 < N) {
            u0 = u[(long long)(t + 1) * 50 + j0];
            u1 = has1 ? u[(long long)(t + 1) * 50 + j1] : 0.f;
        }
        __syncthreads();                // single wave: lowers to wait, no real barrier
        hprev[j0] = h0;
        if (has1) hprev[j1] = h1;
        __syncthreads();
    }
}

// ---------------- final linear (30x30) + log_softmax ----------------
__global__ void k_lin3_logsoftmax(const float* __restrict__ X, const float* __restrict__ W,
                                  const float* __restrict__ bias, float* __restrict__ out, int N) {
    int n = blockIdx.x * blockDim.x + threadIdx.x;
    if (n >= N) return;
    float xi[30];
#pragma unroll
    for (int k = 0; k < 30; k++) xi[k] = X[(long long)n * 30 + k];
    float z[30];
    float m = -1e30f;
    for (int j = 0; j < 30; j++) {
        float a = bias[j];
#pragma unroll
        for (int k = 0; k < 30; k++) a += xi[k] * W[j * 30 + k];
        z[j] = a;
        m = fmaxf(m, a);
    }
    float s = 0.f;
    for (int j = 0; j < 30; j++) s += expf(z[j] - m);
    float lse = m + logf(s);
    for (int j = 0; j < 30; j++) out[(long long)n * 30 + j] = z[j] - lse;
}

// ---------------- host orchestration ----------------
extern "C" void kernel_launch(void* const* d_in, const int* in_sizes, int n_in,
                              void* d_out, int out_size, void* d_ws, size_t ws_size,
                              hipStream_t stream) {
    const float*     x    = (const float*)d_in[0];
    const long long* ei   = (const long long*)d_in[1];   // int64 in reference
    const float* W1  = (const float*)d_in[2];  const float* b1  = (const float*)d_in[3];
    const float* W2  = (const float*)d_in[4];  const float* b2  = (const float*)d_in[5];
    const float* g1  = (const float*)d_in[6];  const float* be1 = (const float*)d_in[7];
    const float* g2  = (const float*)d_in[8];  const float* be2 = (const float*)d_in[9];
    const float* g3  = (const float*)d_in[10]; const float* be3 = (const float*)d_in[11];
    const float* g4  = (const float*)d_in[12]; const float* be4 = (const float*)d_in[13];
    const float* Wih = (const float*)d_in[14]; const float* Whh = (const float*)d_in[15];
    const float* bih = (const float*)d_in[16]; const float* bhh = (const float*)d_in[17];
    const float* lw1 = (const float*)d_in[18]; const float* lb1 = (const float*)d_in[19];
    const float* lw2 = (const float*)d_in[20]; const float* lb2 = (const float*)d_in[21];
    const float* lw3 = (const float*)d_in[22]; const float* lb3 = (const float*)d_in[23];

    const int N = NNODES;
    const int E = in_sizes[1] / 2;   // 800000

    // workspace carve-out (floats), 256B-aligned slices
    float* ws = (float*)d_ws;
    size_t off = 0;
    auto take = [&](size_t nfloats) { float* p = ws + off; off += (nfloats + 63) & ~(size_t)63; return p; };
    float* dinv  = take((size_t)N);          // degree -> dinv (in place)
    float* bufA  = take((size_t)N * 100);    // GEMM outputs (xw)
    float* bufB  = take((size_t)N * 100);    // aggregated / normalized features
    float* bufC  = take((size_t)N * 50);     // u, then l1
    float* bufD  = take((size_t)N * 50);     // rnn outputs
    float* bufE  = take((size_t)N * 30);     // l2
    float* stats = take(256);
    float* ss    = take(256);
    float* outp  = (float*)d_out;

    const int TB = 256;
    dim3 b256(TB);
    dim3 gN((N + TB - 1) / TB);
    dim3 gE((E + TB - 1) / TB);
    dim3 gNF100((int)(((long long)N * 100 + TB - 1) / TB));
    dim3 gNF50((int)(((long long)N * 50 + TB - 1) / TB));
    dim3 gNF30((int)(((long long)N * 30 + TB - 1) / TB));
    dim3 gEdge(32768);
    dim3 wblk(32);

    // ---- degrees (self-loops included) -> dinv ----
    k_fill<<<gN, b256, 0, stream>>>(dinv, 1.0f, N);
    k_deg_scatter<<<gE, b256, 0, stream>>>(ei, dinv, E);
    k_rsqrt_inplace<<<gN, b256, 0, stream>>>(dinv, N);

    // ---- GCN layer 1 ----
    k_wmma_gemm<100, 100, 0><<<dim3(N / 16, 7), wblk, 0, stream>>>(x, W1, nullptr, nullptr, bufA);
    k_agg_init<<<gNF100, b256, 0, stream>>>(bufA, dinv, b1, bufB, N, 100);
    k_agg_edges<<<gEdge, b256, 0, stream>>>(ei, bufA, dinv, bufB, E, 100);
    k_fill<<<dim3(1), b256, 0, stream>>>(stats, 0.f, 256);
    k_bn_stats<<<dim3(256), dim3(128), 0, stream>>>(bufB, stats, N, 100);
    k_bn_final<<<dim3(1), dim3(128), 0, stream>>>(stats, g1, be1, ss, N, 100);
    k_bn_apply<<<gNF100, b256, 0, stream>>>(bufB, ss, N, 100, 1);

    // ---- GCN layer 2 ----
    k_wmma_gemm<100, 100, 0><<<dim3(N / 16, 7), wblk, 0, stream>>>(bufB, W2, nullptr, nullptr, bufA);
    k_agg_init<<<gNF100, b256, 0, stream>>>(bufA, dinv, b2, bufB, N, 100);
    k_agg_edges<<<gEdge, b256, 0, stream>>>(ei, bufA, dinv, bufB, E, 100);
    k_fill<<<dim3(1), b256, 0, stream>>>(stats, 0.f, 256);
    k_bn_stats<<<dim3(256), dim3(128), 0, stream>>>(bufB, stats, N, 100);
    k_bn_final<<<dim3(1), dim3(128), 0, stream>>>(stats, g2, be2, ss, N, 100);
    k_bn_apply<<<gNF100, b256, 0, stream>>>(bufB, ss, N, 100, 0);

    // ---- RNN: u = h @ Wih^T + bih + bhh, then serial scan ----
    k_wmma_gemm<100, 50, 1><<<dim3(N / 16, 4), wblk, 0, stream>>>(bufB, Wih, bih, bhh, bufC);
    k_rnn_scan<<<dim3(1), dim3(32), 0, stream>>>(bufC, Whh, bufD, N);

    // ---- linear 1 + BN3 + relu ----
    k_wmma_gemm<50, 50, 1><<<dim3(N / 16, 4), wblk, 0, stream>>>(bufD, lw1, lb1, nullptr, bufC);
    k_fill<<<dim3(1), b256, 0, stream>>>(stats, 0.f, 256);
    k_bn_stats<<<dim3(256), dim3(128), 0, stream>>>(bufC, stats, N, 50);
    k_bn_final<<<dim3(1), dim3(128), 0, stream>>>(stats, g3, be3, ss, N, 50);
    k_bn_apply<<<gNF50, b256, 0, stream>>>(bufC, ss, N, 50, 1);

    // ---- linear 2 + BN4 + relu ----
    k_wmma_gemm<50, 30, 1><<<dim3(N / 16, 2), wblk, 0, stream>>>(bufC, lw2, lb2, nullptr, bufE);
    k_fill<<<dim3(1), b256, 0, stream>>>(stats, 0.f, 256);
    k_bn_stats<<<dim3(256), dim3(128), 0, stream>>>(bufE, stats, N, 30);
    k_bn_final<<<dim3(1), dim3(128), 0, stream>>>(stats, g4, be4, ss, N, 30);
    k_bn_apply<<<gNF30, b256, 0, stream>>>(bufE, ss, N, 30, 1);

    // ---- linear 3 + log_softmax -> d_out ----
    k_lin3_logsoftmax<<<gN, b256, 0, stream>>>(bufE, lw3, lb3, outp, N);
}